// EdgeConv_28518582845515
// MI455X (gfx1250) — compile-verified
//
#include <hip/hip_runtime.h>
#include <hip/hip_bf16.h>

typedef __attribute__((ext_vector_type(2))) float v2f;
typedef __attribute__((ext_vector_type(8))) float v8f;

#define BATCH 4
#define CH    64
#define NPTS  4096
#define OCH   64
#define KNN   20
#define CNT_F (4.0f * 4096.0f * 20.0f)

__device__ __forceinline__ v8f wmma_f32_16x16x4(v2f a, v2f b, v8f c) {
    // D = A(16x4 f32) * B(4x16 f32) + C ; full f32 matrix pipe
    return __builtin_amdgcn_wmma_f32_16x16x4_f32(false, a, false, b, (short)0, c,
                                                 false, false);
}

// Async copy 16B global -> LDS (ASYNCcnt path, no VGPR staging).
__device__ __forceinline__ void async_copy_b128(unsigned lds_addr, const void* g) {
    asm volatile("global_load_async_to_lds_b128 %0, %1, off"
                 :: "v"(lds_addr), "v"(g) : "memory");
}
__device__ __forceinline__ void wait_async0() {
    asm volatile("s_wait_asynccnt 0" ::: "memory");
}

// ---------------------------------------------------------------- Kernel A
// xx[b,n] = sum_c x[b,c,n]^2 ; also zero the BN stat accumulators.
__global__ __launch_bounds__(256) void norms_kernel(const float* __restrict__ x,
                                                    float* __restrict__ xx,
                                                    float* __restrict__ ssum,
                                                    float* __restrict__ ssq) {
    if (blockIdx.x == 0 && threadIdx.x < OCH) {
        ssum[threadIdx.x] = 0.f;
        ssq[threadIdx.x]  = 0.f;
    }
    const int t = blockIdx.x * 256 + threadIdx.x;   // 0 .. B*N-1
    const int b = t >> 12, n = t & (NPTS - 1);
    const float* xb = x + (size_t)b * CH * NPTS + n;
    float s = 0.f;
#pragma unroll
    for (int c = 0; c < CH; ++c) {
        float v = xb[c * NPTS];
        s += v * v;
    }
    xx[t] = s;
}

// ---------------------------------------------------------------- Kernel B
// Fused Gram-matrix (WMMA f32 16x16x4) + per-row top-20 selection.
// Single-wave workgroup (barriers degrade to S_NOP) owning 16 rows.
__global__ __launch_bounds__(32) void knn_kernel(const float* __restrict__ x,
                                                 const float* __restrict__ xx,
                                                 int* __restrict__ idx) {
    __shared__ float dtile[16][17];   // padded vs bank conflicts
    __shared__ float topv[16][KNN];
    __shared__ int   topi[16][KNN];
    __shared__ float xxs[16];

    const int lane = threadIdx.x & 31;
    const int m    = lane & 15;          // row/col within tile
    const int hh   = lane >> 4;          // K-half selector for f32 wmma frags

    const int rowTile = blockIdx.x;                   // 0..1023
    const int b  = rowTile >> 8;                      // 256 tiles per batch
    const int n0 = (rowTile & 255) * 16;
    const float* xb  = x + (size_t)b * CH * NPTS;
    const float* xxb = xx + b * NPTS;

    // Preload the A fragments for all 16 K-steps (row block reused 256x).
    v2f afrag[16];
#pragma unroll
    for (int kk = 0; kk < 16; ++kk) {
        const int c = kk * 4 + 2 * hh;
        afrag[kk].x = xb[(size_t)c * NPTS + n0 + m];
        afrag[kk].y = xb[(size_t)(c + 1) * NPTS + n0 + m];
    }

    if (lane < 16) {
#pragma unroll
        for (int t = 0; t < KNN; ++t) {
            topv[lane][t] = -3.0e38f;
            topi[lane][t] = 0;
        }
    }
    float minv = -3.0e38f;
    int   minp = 0;
    const float xxi = xxb[n0 + m];

    for (int m0 = 0; m0 < NPTS; m0 += 16) {
        if (m0 + 16 < NPTS)   // prefetch next column tile
            __builtin_prefetch(xb + (size_t)(2 * hh) * NPTS + m0 + 16 + m, 0, 1);

        v8f acc = {0.f, 0.f, 0.f, 0.f, 0.f, 0.f, 0.f, 0.f};
#pragma unroll
        for (int kk = 0; kk < 16; ++kk) {
            const int c = kk * 4 + 2 * hh;
            v2f bfr;
            bfr.x = xb[(size_t)c * NPTS + m0 + m];
            bfr.y = xb[(size_t)(c + 1) * NPTS + m0 + m];
            acc = wmma_f32_16x16x4(afrag[kk], bfr, acc);
        }
        // Dump D tile: lane holds column m, rows v + 8*hh.
#pragma unroll
        for (int v = 0; v < 8; ++v) dtile[v + 8 * hh][m] = acc[v];
        if (lane < 16) xxs[lane] = xxb[m0 + lane];
        __syncthreads();   // single-wave WG: S_NOP + LDS ordering

        if (lane < 16) {
            // pd = 2*dot - ||xn||^2 - ||xm||^2  (maximize == nearest)
            for (int j = 0; j < 16; ++j) {
                const float pd = 2.f * dtile[lane][j] - xxi - xxs[j];
                if (pd > minv) {
                    topv[lane][minp] = pd;
                    topi[lane][minp] = m0 + j;
                    minv = topv[lane][0];
                    minp = 0;
#pragma unroll
                    for (int t = 1; t < KNN; ++t) {
                        const float tv = topv[lane][t];
                        if (tv < minv) { minv = tv; minp = t; }
                    }
                }
            }
        }
        __syncthreads();
    }

    if (lane < 16) {
        const int base = ((b * NPTS) + n0 + lane) * KNN;
#pragma unroll
        for (int t = 0; t < KNN; ++t) idx[base + t] = topi[lane][t];
    }
}

// ---------------------------------------------------------------- Kernel C/E
// h[b,n,k,o] = edge(k,:) . W(o,:) via WMMA f32 16x16x4, M = k rows (20, padded)
// A fragments loaded once per K-step and reused across 4 o-tiles and both
// m-tiles (8 independent WMMAs back-to-back, acc[2][4]).
// STATS=true : accumulate sum/sumsq per channel into global atomics.
// STATS=false: affine BN + leaky relu + max over k -> out[b,o,n].
template <bool STATS>
__global__ __launch_bounds__(64) void edge_kernel(const float* __restrict__ x,
                                                  const float* __restrict__ W,
                                                  const int* __restrict__ idx,
                                                  const float* __restrict__ bnscale,
                                                  const float* __restrict__ bnbias,
                                                  float* __restrict__ ssum,
                                                  float* __restrict__ ssq,
                                                  float* __restrict__ out) {
    __shared__ __attribute__((aligned(16))) float wlds[OCH * 2 * CH]; // 32 KB
    __shared__ float eds[2][KNN * 2 * CH];                            // 20 KB
    __shared__ float clds[2][CH];
    __shared__ int   jlds[2][KNN];

    const int tid  = threadIdx.x;
    const int wave = tid >> 5;
    const int lane = tid & 31;
    const int m    = lane & 15;
    const int hh   = lane >> 4;

    // Stage W (64x128 f32) into LDS with async global->LDS b128 copies.
    {
        const unsigned ldsbase = (unsigned)(uintptr_t)(&wlds[0]);
        for (int t = tid; t < (OCH * 2 * CH) / 4; t += 64)
            async_copy_b128(ldsbase + t * 16, (const char*)W + t * 16);
        wait_async0();
    }
    __syncthreads();

    float sacc[4]  = {0.f, 0.f, 0.f, 0.f};
    float s2acc[4] = {0.f, 0.f, 0.f, 0.f};

    const int gw = blockIdx.x * 2 + wave;    // 2048 waves, 8 points each
    for (int p = 0; p < 8; ++p) {
        const int pt = gw * 8 + p;
        const int b = pt >> 12, n = pt & (NPTS - 1);
        const float* xb = x + (size_t)b * CH * NPTS;

        clds[wave][lane]      = xb[(size_t)lane * NPTS + n];
        clds[wave][lane + 32] = xb[(size_t)(lane + 32) * NPTS + n];
        if (lane < KNN) jlds[wave][lane] = idx[(b * NPTS + n) * KNN + lane];
        __syncthreads();

        // edge[k, c] = (c<64) ? x[b,c,j_k] - center[c] : center[c-64]
        for (int t = lane; t < KNN * 2 * CH; t += 32) {
            const int k = t >> 7, c = t & 127;
            const float v = (c < CH)
                                ? (xb[(size_t)c * NPTS + jlds[wave][k]] - clds[wave][c])
                                : clds[wave][c - CH];
            eds[wave][t] = v;
        }
        __syncthreads();

        float scale[4], bias[4], omax[4];
        if (!STATS) {
#pragma unroll
            for (int ot = 0; ot < 4; ++ot) {
                scale[ot] = bnscale[ot * 16 + m];
                bias[ot]  = bnbias[ot * 16 + m];
                omax[ot]  = -3.0e38f;
            }
        }

        v8f acc[2][4];
#pragma unroll
        for (int mt = 0; mt < 2; ++mt)
#pragma unroll
            for (int ot = 0; ot < 4; ++ot)
                acc[mt][ot] = (v8f){0.f, 0.f, 0.f, 0.f, 0.f, 0.f, 0.f, 0.f};

        const int   kc1   = (16 + m < KNN) ? 16 + m : 0;   // m-tile 1 row
        const float mask1 = (16 + m < KNN) ? 1.f : 0.f;    // zero pad rows
#pragma unroll
        for (int kk = 0; kk < 32; ++kk) {
            const int c0 = kk * 4 + 2 * hh;
            v2f a0, a1;
            a0.x = eds[wave][m * 128 + c0];
            a0.y = eds[wave][m * 128 + c0 + 1];
            a1.x = eds[wave][kc1 * 128 + c0] * mask1;
            a1.y = eds[wave][kc1 * 128 + c0 + 1] * mask1;
#pragma unroll
            for (int ot = 0; ot < 4; ++ot) {
                const int o = ot * 16 + m;
                v2f bb;
                bb.x = wlds[o * 128 + c0];         // B[c,o] = W[o,c]
                bb.y = wlds[o * 128 + c0 + 1];
                acc[0][ot] = wmma_f32_16x16x4(a0, bb, acc[0][ot]);
                acc[1][ot] = wmma_f32_16x16x4(a1, bb, acc[1][ot]);
            }
        }

#pragma unroll
        for (int mt = 0; mt < 2; ++mt) {
#pragma unroll
            for (int ot = 0; ot < 4; ++ot) {
#pragma unroll
                for (int v = 0; v < 8; ++v) {
                    const int kg = mt * 16 + v + 8 * hh;   // global k of acc[v]
                    if (kg < KNN) {
                        const float hv = acc[mt][ot][v];
                        if (STATS) {
                            sacc[ot]  += hv;
                            s2acc[ot] += hv * hv;
                        } else {
                            float hn = hv * scale[ot] + bias[ot];
                            hn = (hn >= 0.f) ? hn : 0.2f * hn;
                            omax[ot] = fmaxf(omax[ot], hn);
                        }
                    }
                }
            }
        }
        if (!STATS) {
#pragma unroll
            for (int ot = 0; ot < 4; ++ot) {
                const float mm = fmaxf(omax[ot], __shfl_xor(omax[ot], 16, 32));
                if (lane < 16)
                    out[(size_t)b * OCH * NPTS + (size_t)(ot * 16 + lane) * NPTS + n] = mm;
            }
        }
        __syncthreads();   // eds/clds reused next point
    }

    if (STATS) {
#pragma unroll
        for (int ot = 0; ot < 4; ++ot) {
            const float s  = sacc[ot]  + __shfl_xor(sacc[ot], 16, 32);
            const float s2 = s2acc[ot] + __shfl_xor(s2acc[ot], 16, 32);
            if (lane < 16) {
                atomicAdd(&ssum[ot * 16 + lane], s);
                atomicAdd(&ssq[ot * 16 + lane], s2);
            }
        }
    }
}

// ---------------------------------------------------------------- Kernel D
__global__ __launch_bounds__(64) void finalize_kernel(const float* __restrict__ ssum,
                                                      const float* __restrict__ ssq,
                                                      const float* __restrict__ gamma,
                                                      const float* __restrict__ beta,
                                                      float* __restrict__ bnscale,
                                                      float* __restrict__ bnbias) {
    const int o = threadIdx.x;
    if (o < OCH) {
        const float mean = ssum[o] / CNT_F;
        const float var  = ssq[o] / CNT_F - mean * mean;
        const float sc   = gamma[o] * rsqrtf(var + 1e-5f);
        bnscale[o] = sc;
        bnbias[o]  = beta[o] - mean * sc;
    }
}

// ---------------------------------------------------------------- launcher
extern "C" void kernel_launch(void* const* d_in, const int* in_sizes, int n_in,
                              void* d_out, int out_size, void* d_ws, size_t ws_size,
                              hipStream_t stream) {
    const float* x     = (const float*)d_in[0];   // (4, 64, 4096)
    const float* W     = (const float*)d_in[1];   // (64, 128)
    const float* gamma = (const float*)d_in[2];   // (64,)
    const float* beta  = (const float*)d_in[3];   // (64,)
    float* out = (float*)d_out;                   // (4, 64, 4096)

    char* ws = (char*)d_ws;
    float* xx      = (float*)ws;                                    // 16384 f
    int*   idx     = (int*)(ws + 65536);                            // 327680 i
    float* ssum    = (float*)(ws + 65536 + 1310720);                // 64 f
    float* ssq     = ssum + 64;
    float* bnscale = ssum + 128;
    float* bnbias  = ssum + 192;

    norms_kernel<<<64, 256, 0, stream>>>(x, xx, ssum, ssq);
    knn_kernel<<<1024, 32, 0, stream>>>(x, xx, idx);
    edge_kernel<true><<<1024, 64, 0, stream>>>(x, W, idx, nullptr, nullptr,
                                               ssum, ssq, nullptr);
    finalize_kernel<<<1, 64, 0, stream>>>(ssum, ssq, gamma, beta, bnscale, bnbias);
    edge_kernel<false><<<1024, 64, 0, stream>>>(x, W, idx, bnscale, bnbias,
                                                nullptr, nullptr, out);
}